// ResidualVectorQuantizer_84293028151914
// MI455X (gfx1250) — compile-verified
//
#include <hip/hip_runtime.h>

typedef float v2f __attribute__((ext_vector_type(2)));
typedef float v8f __attribute__((ext_vector_type(8)));
typedef unsigned int u32x4 __attribute__((ext_vector_type(4)));
typedef int i32x4 __attribute__((ext_vector_type(4)));
typedef int i32x8 __attribute__((ext_vector_type(8)));

#define DIM       128
#define NCB       8
#define KCB       2048
#define BATCH     16
#define TOKENS    2048
#define ROWW      132          // padded LDS row width in floats (128 + 2x2 pad)
#define NTILES    (KCB / 16)   // 128 codeword tiles per stage

// ---------------------------------------------------------------------------
// Kernel 1: zero the loss accumulator slot in d_out.
// ---------------------------------------------------------------------------
__global__ void rvq_zero_loss(float* loss) { *loss = 0.0f; }

// ---------------------------------------------------------------------------
// Kernel 2: precompute ||c_k||^2 for every codeword of every codebook.
// ---------------------------------------------------------------------------
__global__ void rvq_cnorm(const float* __restrict__ codebooks,
                          float* __restrict__ cnorm) {
  int i = blockIdx.x * blockDim.x + threadIdx.x;
  if (i >= NCB * KCB) return;
  const float* c = codebooks + (size_t)i * DIM;
  float s = 0.0f;
#pragma unroll 8
  for (int d = 0; d < DIM; ++d) s += c[d] * c[d];
  cnorm[i] = s;
}

// ---------------------------------------------------------------------------
// TDM: DMA one 16x128 f32 codebook tile into LDS with row padding.
// D# group1: data_size=4B (2), pad_enable, pad_interval code 5 (=64 DWORDs),
// pad_amount code 1 (=2 DWORDs) -> each 128-float row lands on 132 words,
// giving a 4-bank shift per row (conflict-free b64 reads by 32 lanes).
// ---------------------------------------------------------------------------
__device__ __forceinline__ void tdm_load_tile(float* lds_dst,
                                              const float* gsrc) {
  const unsigned lds_off = (unsigned)(unsigned long long)(uintptr_t)lds_dst;
  const unsigned long long ga = (unsigned long long)(uintptr_t)gsrc;

  u32x4 g0 = { 1u,                                        // count=1, user D#
               lds_off,                                   // lds_addr (bytes)
               (unsigned)(ga & 0xffffffffull),            // global_addr[31:0]
               (unsigned)((ga >> 32) & 0x01ffffffull)     // global_addr[56:32]
                   | (2u << 30) };                        // type=2 ("image")

  i32x8 g1 = { (int)((2u << 16)     // data_size = 4 bytes
                   | (1u << 20)     // pad_enable
                   | (5u << 22)     // pad_interval: 64 DWORDs
                   | (1u << 25)),   // pad_amount:   2 DWORDs
               (int)(128u << 16),                  // tensor_dim0 lo = 128
               (int)(16u << 16),                   // dim0 hi=0 | tensor_dim1 lo = 16
               (int)(128u << 16),                  // dim1 hi=0 | tile_dim0 = 128
               (int)16,                            // tile_dim1 = 16 | tile_dim2 = 0
               (int)128,                           // tensor_dim0_stride lo = 128
               0, 0 };                             // stride0 hi | stride1 = 0

  i32x4 z4 = { 0, 0, 0, 0 };
#if __has_include(<hip/amd_detail/amd_gfx1250_TDM.h>)
  i32x8 z8 = { 0, 0, 0, 0, 0, 0, 0, 0 };
  __builtin_amdgcn_tensor_load_to_lds(g0, g1, z4, z4, z8, 0);
#else
  __builtin_amdgcn_tensor_load_to_lds(g0, g1, z4, z4, 0);
#endif
}

// ---------------------------------------------------------------------------
// Kernel 3: fused RVQ forward. One wave owns 16 tokens (one WMMA M-tile).
// Residual in registers (f32 16x4 A-fragment layout); codebook tiles staged
// in LDS by the Tensor Data Mover, double-buffered, shared by all 4 waves.
// ---------------------------------------------------------------------------
__global__ __launch_bounds__(128)
void rvq_main(const float* __restrict__ x,
              const float* __restrict__ codebooks,
              const float* __restrict__ cnorms,
              float* __restrict__ out_total,
              float* __restrict__ out_codes,
              float* __restrict__ out_loss) {
  const int lane = threadIdx.x & 31;
  const int wave = threadIdx.x >> 5;
  const int tile = blockIdx.x * 4 + wave;      // 0 .. 2047 token tiles
  const int b    = tile >> 7;
  const int tb   = (tile & 127) << 4;
  const int row  = lane & 15;                  // token row / codeword column
  const int hi   = lane >> 4;                  // dim sub-pair selector
  const int t    = tb + row;

  __shared__ float s_tile[2][16 * ROWW];       // double-buffered codebook tile
  __shared__ int   s_idx[4][16];

  // ---- load residual fragments (x is (B, D, T): dim stride = TOKENS) ----
  const float* xb = x + (size_t)b * DIM * TOKENS + t;
  float r0[32], r1[32];
#pragma unroll
  for (int kk = 0; kk < 32; ++kk) {
    const int d = 4 * kk + 2 * hi;
    r0[kk] = xb[(size_t)d * TOKENS];
    r1[kk] = xb[(size_t)(d + 1) * TOKENS];
  }

  float loss_acc = 0.0f;

  for (int cb = 0; cb < NCB; ++cb) {
    const float* C  = codebooks + (size_t)cb * KCB * DIM;
    const float* CN = cnorms + cb * KCB;

    float minv[8];
    int   mini[8];
#pragma unroll
    for (int s = 0; s < 8; ++s) { minv[s] = 3.4e38f; mini[s] = 0; }

    // prefetch tile 0
    if (wave == 0) tdm_load_tile(&s_tile[0][0], C);

    for (int it = 0; it < NTILES; ++it) {
      const int buf = it & 1;
      __syncthreads();              // prior compute done -> buf^1 reusable
      if (wave == 0) {
        if (it + 1 < NTILES) {
          tdm_load_tile(&s_tile[buf ^ 1][0], C + (size_t)(it + 1) * 16 * DIM);
          __builtin_amdgcn_s_wait_tensorcnt(1);   // tile `it` complete
        } else {
          __builtin_amdgcn_s_wait_tensorcnt(0);
        }
      }
      __syncthreads();              // tile `it` visible to all waves

      const int    kb   = it * 16;
      const int    col  = kb + row;
      const float* crow = &s_tile[buf][ROWW * row + 2 * hi];

      v8f acc0 = {};
      v8f acc1 = {};
      // two independent WMMA chains (hazard hiding); B from LDS (ds_load_b64)
#pragma unroll
      for (int kk = 0; kk < 16; ++kk) {
        v2f a  = { r0[kk], r1[kk] };
        v2f bb = *(const v2f*)(crow + 4 * kk);          // cols < 64: no pad
        acc0 = __builtin_amdgcn_wmma_f32_16x16x4_f32(
            false, a, false, bb, (short)0, acc0, false, false);
      }
#pragma unroll
      for (int kk = 16; kk < 32; ++kk) {
        v2f a  = { r0[kk], r1[kk] };
        v2f bb = *(const v2f*)(crow + 4 * kk + 2);      // +2: skip mid-row pad
        acc1 = __builtin_amdgcn_wmma_f32_16x16x4_f32(
            false, a, false, bb, (short)0, acc1, false, false);
      }

      const float cn = CN[col];
#pragma unroll
      for (int s = 0; s < 8; ++s) {
        const float d = cn - 2.0f * (acc0[s] + acc1[s]);
        if (d < minv[s]) { minv[s] = d; mini[s] = col; }
      }
    }

    // ---- argmin across the 16 column-lanes of each half ----
#pragma unroll
    for (int m = 1; m <= 8; m <<= 1) {
#pragma unroll
      for (int s = 0; s < 8; ++s) {
        const float ov = __shfl_xor(minv[s], m, 32);
        const int   oi = __shfl_xor(mini[s], m, 32);
        if (ov < minv[s] || (ov == minv[s] && oi < mini[s])) {
          minv[s] = ov; mini[s] = oi;
        }
      }
    }

    if (lane == 0) {
#pragma unroll
      for (int s = 0; s < 8; ++s) s_idx[wave][s] = mini[s];
    }
    if (lane == 16) {
#pragma unroll
      for (int s = 0; s < 8; ++s) s_idx[wave][8 + s] = mini[s];
    }
    __syncthreads();
    const int myidx = s_idx[wave][row];

    if (lane < 16) {
      out_codes[((size_t)b * NCB + cb) * TOKENS + t] = (float)myidx;
    }
    __syncthreads();  // s_idx reused next stage

    // ---- residual update + streamed commitment-loss accumulation ----
    const float* q = C + (size_t)myidx * DIM + 2 * hi;
#pragma unroll
    for (int kk = 0; kk < 32; ++kk) {
      v2f qq = *(const v2f*)(q + 4 * kk);
      r0[kk] -= qq.x;
      r1[kk] -= qq.y;
      loss_acc += r0[kk] * r0[kk] + r1[kk] * r1[kk];
    }
  }

  // ---- total_quantized = x - residual_final (identity), layout (B, D, T) ----
  float* ob = out_total + (size_t)b * DIM * TOKENS + t;
#pragma unroll
  for (int kk = 0; kk < 32; ++kk) {
    const int d = 4 * kk + 2 * hi;
    ob[(size_t)d * TOKENS]       = xb[(size_t)d * TOKENS]       - r0[kk];
    ob[(size_t)(d + 1) * TOKENS] = xb[(size_t)(d + 1) * TOKENS] - r1[kk];
  }

  // ---- loss: wave reduction then one atomicAdd ----
#pragma unroll
  for (int m = 1; m <= 16; m <<= 1) loss_acc += __shfl_xor(loss_acc, m, 32);
  if (lane == 0) {
    const float inv_n = 1.0f / ((float)BATCH * (float)TOKENS * (float)DIM);
    atomicAdd(out_loss, loss_acc * inv_n);
  }
}

// ---------------------------------------------------------------------------
extern "C" void kernel_launch(void* const* d_in, const int* in_sizes, int n_in,
                              void* d_out, int out_size, void* d_ws,
                              size_t ws_size, hipStream_t stream) {
  const float* x         = (const float*)d_in[0];  // (16, 128, 2048)
  const float* codebooks = (const float*)d_in[1];  // (8, 2048, 128)

  float* out_total = (float*)d_out;                                // 4194304
  float* out_codes = out_total + (size_t)BATCH * DIM * TOKENS;     //  262144
  float* out_loss  = out_codes + (size_t)BATCH * NCB * TOKENS;     //       1

  float* cnorm = (float*)d_ws;  // NCB*KCB floats = 64 KB

  rvq_zero_loss<<<1, 1, 0, stream>>>(out_loss);

  rvq_cnorm<<<(NCB * KCB + 255) / 256, 256, 0, stream>>>(codebooks, cnorm);

  rvq_main<<<512, 128, 0, stream>>>(x, codebooks, cnorm,
                                    out_total, out_codes, out_loss);
}